// OIMUnsupervisedLoss_32916629357082
// MI455X (gfx1250) — compile-verified
//
#include <hip/hip_runtime.h>
#include <hip/hip_bf16.h>

// ---------------- problem constants ----------------
#define NFEAT   256
#define NROWS   4096          // B_TOTAL
#define NPID    5000          // rows of lut_instance / reid_lut
#define LDNPAD  5056          // NPID padded to 64 (79 * 64)
#define MTILES  64            // 4096 / 64
#define NTILES  79            // ceil(5056 / 64)
#define OIM_SCALE 30.0f
#define IGNORE_INDEX 5554

typedef __attribute__((ext_vector_type(2))) float v2f;
typedef __attribute__((ext_vector_type(8))) float v8f;

// accumulator slots in workspace
#define ACC_FC   0
#define ACC_N1   1
#define ACC_N2   2
#define ACC_N3   3
#define ACC_INST 4
#define ACC_CE   5

// ---------------- block reductions (blockDim.x == 256) ----------------
__device__ __forceinline__ float blockSum256(float v, float* sm) {
  int tid = threadIdx.x;
  sm[tid] = v;
  __syncthreads();
  #pragma unroll
  for (int s = 128; s >= 1; s >>= 1) {
    if (tid < s) sm[tid] += sm[tid + s];
    __syncthreads();
  }
  float r = sm[0];
  __syncthreads();
  return r;
}

__device__ __forceinline__ float blockMax256(float v, float* sm) {
  int tid = threadIdx.x;
  sm[tid] = v;
  __syncthreads();
  #pragma unroll
  for (int s = 128; s >= 1; s >>= 1) {
    if (tid < s) sm[tid] = fmaxf(sm[tid], sm[tid + s]);
    __syncthreads();
  }
  float r = sm[0];
  __syncthreads();
  return r;
}

// ---------------- kernel 0: zero the scalar accumulators ----------------
__global__ void __launch_bounds__(32) oim_zero_acc(float* acc) {
  if (threadIdx.x < 8) acc[threadIdx.x] = 0.0f;
}

// ---------------- kernel 1: per-row prep + masked MSE ----------------
// grid = NROWS blocks, 256 threads (one thread per feature)
__global__ void __launch_bounds__(256)
oim_prep(const float* __restrict__ inputs,
         const int*   __restrict__ roi_label,
         const float* __restrict__ lut_instance,
         const int*   __restrict__ reid_labels,
         int* __restrict__ lr_arr,
         int* __restrict__ w2_arr,
         int* __restrict__ w3_arr,
         float* __restrict__ acc) {
  __shared__ float sm[256];
  int row = blockIdx.x;
  int tid = threadIdx.x;

  int t      = roi_label[row] - 1;
  int valid  = (t >= 0);
  int label  = valid ? t : 0;
  int lrr    = reid_labels[label];
  int valid2 = valid && (lrr >= 0);
  int lr     = (lrr >= 0) ? lrr : 0;
  int valid3 = valid2 && (lr != IGNORE_INDEX);

  float d  = lut_instance[(size_t)label * NFEAT + tid] - inputs[(size_t)row * NFEAT + tid];
  float ss = blockSum256(d * d, sm);

  if (tid == 0) {
    lr_arr[row] = lr;
    w2_arr[row] = valid2;
    w3_arr[row] = valid3;
    if (valid)  { atomicAdd(&acc[ACC_FC], ss); atomicAdd(&acc[ACC_N1], 1.0f); }
    if (valid2) atomicAdd(&acc[ACC_N2], 1.0f);
    if (valid3) atomicAdd(&acc[ACC_N3], 1.0f);
  }
}

// ---------------- kernel 2: f32 WMMA GEMM: Out = scale * A @ Bm^T ----------------
// A: [NROWS x NFEAT] row-major, Bm: [NPID x NFEAT] row-major,
// Out: [NROWS x LDNPAD]. One wave computes a 64x64 tile (4x4 WMMA fragments).
// __launch_bounds__(256): only 8 waves/WGP must fit -> full VGPR budget, no
// accumulator spilling (16 x v8f = 128 VGPRs stay register-resident).
__global__ void __launch_bounds__(256)
oim_gemm_wmma(const float* __restrict__ A,
              const float* __restrict__ Bm,
              float* __restrict__ Out,
              float scale) {
  int wave = (int)((blockIdx.x * blockDim.x + threadIdx.x) >> 5);
  int lane = threadIdx.x & 31;
  int half = lane >> 4;   // 0 or 1
  int idx  = lane & 15;   // 0..15

  int mTile = wave % MTILES;
  int nTile = wave / MTILES;
  if (nTile >= NTILES) return;
  int mBase = mTile * 64;
  int nBase = nTile * 64;

  // A fragment element: row = mBase+16*i+idx, K = kk + 2*half + v
  // Single base pointer; the i*16-row hop (16384 B) folds into the 24-bit
  // instruction offset, keeping address-register pressure low.
  const float* aBase = A + (size_t)(mBase + idx) * NFEAT + 2 * half;

  // B fragment element: row of Bm = nBase+16*j+idx (clamped), same K pattern.
  const float* bPtr[4];
  #pragma unroll
  for (int j = 0; j < 4; ++j) {
    int r = nBase + 16 * j + idx;
    if (r > NPID - 1) r = NPID - 1;           // clamp; padded cols ignored later
    bPtr[j] = Bm + (size_t)r * NFEAT + 2 * half;
  }

  v8f zero = {};
  v8f accf[4][4];
  #pragma unroll
  for (int i = 0; i < 4; ++i)
    #pragma unroll
    for (int j = 0; j < 4; ++j) accf[i][j] = zero;

  for (int kk = 0; kk < NFEAT; kk += 4) {
    v2f af[4], bf[4];
    #pragma unroll
    for (int i = 0; i < 4; ++i)
      af[i] = *(const v2f*)(aBase + (size_t)i * 16 * NFEAT + kk);
    #pragma unroll
    for (int j = 0; j < 4; ++j)
      bf[j] = *(const v2f*)(bPtr[j] + kk);
    #pragma unroll
    for (int i = 0; i < 4; ++i) {
      #pragma unroll
      for (int j = 0; j < 4; ++j) {
        accf[i][j] = __builtin_amdgcn_wmma_f32_16x16x4_f32(
            false, af[i], false, bf[j], (short)0, accf[i][j], false, false);
      }
    }
  }

  // D element: vgpr r, lane -> M = 16*i + r + 8*half, N = 16*j + idx
  #pragma unroll
  for (int i = 0; i < 4; ++i) {
    #pragma unroll
    for (int j = 0; j < 4; ++j) {
      #pragma unroll
      for (int r = 0; r < 8; ++r) {
        int M = mBase + 16 * i + r + 8 * half;
        int N = nBase + 16 * j + idx;
        Out[(size_t)M * LDNPAD + N] = accf[i][j][r] * scale;
      }
    }
  }
}

// ---------------- kernel 3: instance loss reduction ----------------
// grid = NROWS blocks, 256 threads; logits row is L2-resident.
__global__ void __launch_bounds__(256)
oim_inst_loss(const float* __restrict__ logits,
              const int*   __restrict__ reid_labels,
              const int*   __restrict__ lr_arr,
              const int*   __restrict__ w2_arr,
              float* __restrict__ acc) {
  int row = blockIdx.x;
  if (!w2_arr[row]) return;
  __shared__ float sm[256];
  int tid = threadIdx.x;
  int lr  = lr_arr[row];
  const float* L = logits + (size_t)row * LDNPAD;

  // logsumexp over negatives (reid_labels[p] != lr)
  float mloc = -INFINITY;
  for (int p = tid; p < NPID; p += 256) {
    float v = L[p];
    if (reid_labels[p] != lr) mloc = fmaxf(mloc, v);
  }
  float m = blockMax256(mloc, sm);

  float sloc = 0.0f;
  for (int p = tid; p < NPID; p += 256) {
    float v = L[p];
    if (reid_labels[p] != lr) sloc += expf(v - m);
  }
  float s = blockSum256(sloc, sm);
  float lse = (s > 0.0f) ? (m + logf(s)) : -INFINITY;

  // positives: per_pos = logaddexp(v, lse) - v = max(lse-v,0)+log1p(exp(-|lse-v|))
  float psLoc = 0.0f, pcLoc = 0.0f;
  for (int p = tid; p < NPID; p += 256) {
    if (reid_labels[p] == lr) {
      float v = L[p];
      float d = lse - v;
      float pp = fmaxf(d, 0.0f) + log1pf(expf(-fabsf(d)));
      psLoc += pp;
      pcLoc += 1.0f;
    }
  }
  float ps = blockSum256(psLoc, sm);
  float pc = blockSum256(pcLoc, sm);
  if (tid == 0) atomicAdd(&acc[ACC_INST], ps / fmaxf(pc, 1.0f));
}

// ---------------- kernel 4: reid cross-entropy reduction ----------------
__global__ void __launch_bounds__(256)
oim_ce_loss(const float* __restrict__ logits,
            const int*   __restrict__ lr_arr,
            const int*   __restrict__ w3_arr,
            float* __restrict__ acc) {
  int row = blockIdx.x;
  if (!w3_arr[row]) return;
  __shared__ float sm[256];
  int tid = threadIdx.x;
  const float* L = logits + (size_t)row * LDNPAD;

  float mloc = -INFINITY;
  for (int p = tid; p < NPID; p += 256) mloc = fmaxf(mloc, L[p]);
  float m = blockMax256(mloc, sm);

  float sloc = 0.0f;
  for (int p = tid; p < NPID; p += 256) sloc += expf(L[p] - m);
  float s = blockSum256(sloc, sm);

  if (tid == 0) {
    float lse = m + logf(s);
    atomicAdd(&acc[ACC_CE], lse - L[lr_arr[row]]);
  }
}

// ---------------- kernel 5: finalize scalar ----------------
__global__ void __launch_bounds__(32)
oim_finalize(const float* __restrict__ acc, float* __restrict__ out) {
  if (threadIdx.x != 0) return;
  float n1 = fmaxf(acc[ACC_N1], 1.0f);
  float n2 = fmaxf(acc[ACC_N2], 1.0f);
  float n3 = fmaxf(acc[ACC_N3], 1.0f);
  out[0] = acc[ACC_FC] / (n1 * (float)NFEAT)   // feature_consistency (MSE over rows*feat)
         + acc[ACC_CE] / n3                    // loss_oim_reid
         + acc[ACC_INST] / n2;                 // instance_loss
}

// ---------------- launcher ----------------
extern "C" void kernel_launch(void* const* d_in, const int* in_sizes, int n_in,
                              void* d_out, int out_size, void* d_ws, size_t ws_size,
                              hipStream_t stream) {
  (void)in_sizes; (void)n_in; (void)out_size; (void)ws_size;

  const float* inputs       = (const float*)d_in[0];  // [4096,256]
  const int*   roi_label    = (const int*)  d_in[1];  // [4096]
  const float* lut_instance = (const float*)d_in[2];  // [5000,256]
  const float* reid_lut     = (const float*)d_in[3];  // [5000,256]
  const int*   reid_labels  = (const int*)  d_in[4];  // [5000]
  float*       out          = (float*)d_out;

  // workspace layout
  float* logits = (float*)d_ws;                       // [4096 x 5056] f32 (~83 MB)
  size_t nLog   = (size_t)NROWS * LDNPAD;
  int*   lr_arr = (int*)(logits + nLog);
  int*   w2_arr = lr_arr + NROWS;
  int*   w3_arr = w2_arr + NROWS;
  float* acc    = (float*)(w3_arr + NROWS);           // 8 scalar accumulators

  oim_zero_acc<<<1, 32, 0, stream>>>(acc);

  oim_prep<<<NROWS, 256, 0, stream>>>(inputs, roi_label, lut_instance, reid_labels,
                                      lr_arr, w2_arr, w3_arr, acc);

  const int gemmBlocks = (MTILES * NTILES) / 8;       // 8 waves per 256-thread block
  // logits = 30 * inputs @ lut_instance^T ; then instance-loss reductions
  oim_gemm_wmma<<<gemmBlocks, 256, 0, stream>>>(inputs, lut_instance, logits, OIM_SCALE);
  oim_inst_loss<<<NROWS, 256, 0, stream>>>(logits, reid_labels, lr_arr, w2_arr, acc);

  // reuse the same buffer: logits = 30 * inputs @ reid_lut^T ; then CE reductions
  oim_gemm_wmma<<<gemmBlocks, 256, 0, stream>>>(inputs, reid_lut, logits, OIM_SCALE);
  oim_ce_loss<<<NROWS, 256, 0, stream>>>(logits, lr_arr, w3_arr, acc);

  oim_finalize<<<1, 1, 0, stream>>>(acc, out);
}